// EmergentRiskMetrics_4329327035077
// MI455X (gfx1250) — compile-verified
//
#include <hip/hip_runtime.h>
#include <math.h>

// ---------------------------------------------------------------------------
// EmergentRiskMetrics for MI455X (gfx1250, wave32)
//
// Heavy parts:
//  * Gram matrices S=f(X)^T f(X), K=8192 -> V_WMMA_F32_16X16X4_F32 (3 modes)
//  * Rolling off-diag corr collapsed algebraically: sum_ij r_ij = sum_t(sum_i z)^2
//  * Power iteration for lambda_max of the 128x128 corr matrix
// ---------------------------------------------------------------------------

typedef __attribute__((ext_vector_type(2))) float v2f;
typedef __attribute__((ext_vector_type(8))) float v8f;

#define T_DIM 8192
#define A_DIM 128
#define NW20  (T_DIM - 20)
#define NW10  (T_DIM - 10)

// workspace layout (float offsets)
#define OFF_CSSTD 0                       // 8192
#define OFF_R20   (OFF_CSSTD + T_DIM)     // 8172
#define OFF_R10   (OFF_R20 + NW20)        // 8182
#define OFF_S0    24576                   // 16384  raw gram
#define OFF_S1    40960                   // 16384  sign gram
#define OFF_S2    57344                   // 16384  nonzero gram
#define OFF_CORR  73728                   // 16384
#define OFF_CSUM  90112                   // 128    column sums (raw)
#define OFF_NA    90240                   // 128    nonzero counts per column
#define OFF_SCAL  90368                   // 16 scalars

__device__ __forceinline__ float waveRedAdd(float v) {
  #pragma unroll
  for (int m = 16; m >= 1; m >>= 1) v += __shfl_xor(v, m, 32);
  return v;
}

// --------------------------- cross-sectional std ---------------------------
// one wave per row; 4 elems/lane; wave32 shuffle reduction
__global__ void csstd_kernel(const float* __restrict__ X, float* __restrict__ cs_std) {
  const int tid = threadIdx.x, wid = tid >> 5, lane = tid & 31;
  const int row = blockIdx.x * 8 + wid;
  float s = 0.f, q = 0.f;
  #pragma unroll
  for (int j = 0; j < 4; ++j) {
    float x = X[row * A_DIM + lane + j * 32];
    s += x; q += x * x;
  }
  s = waveRedAdd(s); q = waveRedAdd(q);
  if (lane == 0) {
    float var = (q - s * s / (float)A_DIM) / (float)(A_DIM - 1);
    cs_std[row] = sqrtf(fmaxf(var, 0.f));
  }
}

// ------------------------------ column sums --------------------------------
__global__ void colsums_kernel(const float* __restrict__ X,
                               float* __restrict__ csum, float* __restrict__ na) {
  const int i = threadIdx.x;  // column, 128 threads
  float s0 = 0.f, s2 = 0.f;
  for (int t = 0; t < T_DIM; ++t) {
    float x = X[t * A_DIM + i];
    s0 += x;
    s2 += (x != 0.f) ? 1.f : 0.f;
  }
  csum[i] = s0;
  na[i]   = s2;
}

// --------------------------- WMMA f32 Gram kernel --------------------------
// out[a][b] = sum_t f(X[t][a]) * f(X[t][b]);   MODE: 0=raw 1=sign 2=nonzero
// 64 blocks (8x8 16x16 tiles) x 128 threads (4 waves, each K/4 partition).
template <int MODE>
__global__ void gram_kernel(const float* __restrict__ X, float* __restrict__ out) {
  __shared__ float red[3 * 256];
  const int bid = blockIdx.x;
  const int a0 = (bid & 7) * 16;
  const int b0 = (bid >> 3) * 16;
  const int tid = threadIdx.x, wid = tid >> 5, lane = tid & 31;
  // A-matrix 16x4 f32 layout: lane l holds A[l&15][2*(l>>4)+r] in VGPR r
  const int colA = a0 + (lane & 15);
  const int colB = b0 + (lane & 15);
  const int krow = (lane >> 4) * 2;

  v8f acc = {0.f, 0.f, 0.f, 0.f, 0.f, 0.f, 0.f, 0.f};
  const int kBeg = wid * (T_DIM / 4);
  const int kEnd = kBeg + (T_DIM / 4);
  for (int t = kBeg; t < kEnd; t += 4) {
    float xa0 = X[(t + krow    ) * A_DIM + colA];
    float xa1 = X[(t + krow + 1) * A_DIM + colA];
    float xb0 = X[(t + krow    ) * A_DIM + colB];
    float xb1 = X[(t + krow + 1) * A_DIM + colB];
    if (MODE == 1) {  // sign(x)
      xa0 = (float)((xa0 > 0.f) - (xa0 < 0.f));
      xa1 = (float)((xa1 > 0.f) - (xa1 < 0.f));
      xb0 = (float)((xb0 > 0.f) - (xb0 < 0.f));
      xb1 = (float)((xb1 > 0.f) - (xb1 < 0.f));
    } else if (MODE == 2) {  // nonzero indicator
      xa0 = (xa0 != 0.f) ? 1.f : 0.f;
      xa1 = (xa1 != 0.f) ? 1.f : 0.f;
      xb0 = (xb0 != 0.f) ? 1.f : 0.f;
      xb1 = (xb1 != 0.f) ? 1.f : 0.f;
    }
    v2f a = {xa0, xa1};
    v2f b = {xb0, xb1};
    // D = A(16x4) x B(4x16) + C, full fp32
    acc = __builtin_amdgcn_wmma_f32_16x16x4_f32(
        /*neg_a=*/false, a, /*neg_b=*/false, b,
        /*c_mod=*/(short)0, acc, /*reuse_a=*/false, /*reuse_b=*/false);
  }
  // cross-wave partial-C reduction through LDS
  if (wid > 0) {
    #pragma unroll
    for (int v = 0; v < 8; ++v) red[(wid - 1) * 256 + lane * 8 + v] = acc[v];
  }
  __syncthreads();
  if (wid == 0) {
    #pragma unroll
    for (int v = 0; v < 8; ++v) {
      float s = acc[v];
      s += red[0 * 256 + lane * 8 + v];
      s += red[1 * 256 + lane * 8 + v];
      s += red[2 * 256 + lane * 8 + v];
      const int m = v + ((lane >> 4) << 3);  // C layout: lanes16-31 hold M+8
      const int n = lane & 15;
      out[(a0 + m) * A_DIM + (b0 + n)] = s;
    }
  }
}

// --------------------------- rolling off-diag corr -------------------------
// offdiag(s) = (sum_t (sum_i z_ti)^2 - A) / (A*(A-1)),  z = (x-mu)/d per window
template <int W>
__global__ void rolling_kernel(const float* __restrict__ X, float* __restrict__ out) {
  __shared__ float z[W * A_DIM];
  __shared__ float wpart[4];
  const int s = blockIdx.x;
  const int i = threadIdx.x;  // column
  const int wid = i >> 5, lane = i & 31;

  float xv[W];
  float cs = 0.f, cq = 0.f;
  #pragma unroll
  for (int t = 0; t < W; ++t) {
    float x = X[(s + t) * A_DIM + i];
    xv[t] = x; cs += x; cq += x * x;
  }
  __builtin_prefetch(&X[(s + W) * A_DIM + i], 0, 0);  // global_prefetch_b8
  const float mu = cs / (float)W;
  const float invd = rsqrtf(fmaxf(cq - cs * mu, 1e-30f));
  #pragma unroll
  for (int t = 0; t < W; ++t) z[t * A_DIM + i] = (xv[t] - mu) * invd;
  __syncthreads();

  float sum2 = 0.f;
  for (int t = wid; t < W; t += 4) {
    float p = z[t * A_DIM + lane]      + z[t * A_DIM + lane + 32] +
              z[t * A_DIM + lane + 64] + z[t * A_DIM + lane + 96];
    p = waveRedAdd(p);
    sum2 += p * p;
  }
  if (lane == 0) wpart[wid] = sum2;
  __syncthreads();
  if (i == 0) {
    float tot = wpart[0] + wpart[1] + wpart[2] + wpart[3];
    out[s] = (tot - (float)A_DIM) / ((float)A_DIM * (float)(A_DIM - 1));
  }
}

// ------------------------------ corr build ---------------------------------
__global__ void build_corr_kernel(const float* __restrict__ S0,
                                  const float* __restrict__ csum,
                                  float* __restrict__ corr) {
  const int idx = blockIdx.x * 256 + threadIdx.x;
  const int a = idx >> 7, b = idx & 127;
  const float Tn = (float)T_DIM;
  const float mua = csum[a] / Tn, mub = csum[b] / Tn;
  const float cov = S0[idx] - Tn * mua * mub;
  const float da = sqrtf(fmaxf(S0[a * A_DIM + a] - Tn * mua * mua, 1e-30f));
  const float db = sqrtf(fmaxf(S0[b * A_DIM + b] - Tn * mub * mub, 1e-30f));
  corr[idx] = cov / (da * db);
}

// -------------------- corr / concordance reductions ------------------------
__global__ void reduce_kernel(const float* __restrict__ corr,
                              const float* __restrict__ s1,
                              const float* __restrict__ s2,
                              const float* __restrict__ na,
                              float* __restrict__ scal) {
  __shared__ float r0[256], r1[256], r2[256], r3[256], r4[256];
  const int tid = threadIdx.x;
  float sc = 0.f, sa = 0.f, p1 = 0.f, p2 = 0.f, sn = 0.f;
  for (int idx = tid; idx < A_DIM * A_DIM; idx += 256) {
    float c = corr[idx]; sc += c; sa += fabsf(c);
    p1 += s1[idx]; p2 += s2[idx];
  }
  if (tid < A_DIM) sn = na[tid];
  r0[tid] = sc; r1[tid] = sa; r2[tid] = p1; r3[tid] = p2; r4[tid] = sn;
  __syncthreads();
  for (int off = 128; off >= 1; off >>= 1) {
    if (tid < off) {
      r0[tid] += r0[tid + off]; r1[tid] += r1[tid + off];
      r2[tid] += r2[tid + off]; r3[tid] += r3[tid + off];
      r4[tid] += r4[tid + off];
    }
    __syncthreads();
  }
  if (tid == 0) {
    const float A = (float)A_DIM, Tn = (float)T_DIM;
    scal[3] = (r0[0] - A) / (A * (A - 1.f));         // avg_correlation
    scal[4] = 1.f - r1[0] / (A * A);                 // return_diversity
    // sign-match count: 0.5*P1 + 1.5*P2 + T - n_a - n_b; diag count == T
    float cnt_off = 0.5f * r2[0] + 1.5f * r3[0] + A * A * Tn - 2.f * A * r4[0] - A * Tn;
    scal[5] = cnt_off / (Tn * A * (A - 1.f));        // avg_concordance
  }
}

// ---------------- cs_std / rolling scalar reductions -----------------------
__global__ void scalar_reduce_kernel(const float* __restrict__ cs_std,
                                     const float* __restrict__ r20,
                                     const float* __restrict__ r10,
                                     float* __restrict__ scal) {
  __shared__ float rA[256], rB[256], rC[256];
  const int tid = threadIdx.x;
  float sdisp = 0.f, plock = 0.f, shist = 0.f;
  for (int t = tid; t < T_DIM; t += 256) sdisp += cs_std[t];
  for (int t = tid; t < NW20; t += 256) plock += (r20[t] > 0.7f) ? 1.f : 0.f;
  for (int t = tid; t < NW10 - 5; t += 256) shist += r10[t];
  rA[tid] = sdisp; rB[tid] = plock; rC[tid] = shist;
  __syncthreads();
  for (int off = 128; off >= 1; off >>= 1) {
    if (tid < off) { rA[tid] += rA[tid + off]; rB[tid] += rB[tid + off]; rC[tid] += rC[tid + off]; }
    __syncthreads();
  }
  if (tid == 0) {
    scal[0] = rA[0] / (float)T_DIM;         // avg_dispersion
    scal[1] = cs_std[0];
    scal[2] = cs_std[T_DIM - 1];
    scal[7] = rB[0] / (float)NW20;          // phase_locking
    scal[9] = rC[0] / (float)(NW10 - 5);    // historical
    float rec = 0.f;
    for (int k = NW10 - 5; k < NW10; ++k) rec += r10[k];
    scal[8] = rec / 5.f;                    // recent
  }
}

// ------------------------------ power iteration ----------------------------
__global__ void power_kernel(const float* __restrict__ corr, float* __restrict__ scal) {
  __shared__ float v[128];
  __shared__ float red[128];
  const int i = threadIdx.x;
  v[i] = 1.f;
  __syncthreads();
  float nrm = 1.f;
  for (int it = 0; it < 96; ++it) {
    float y = 0.f;
    for (int j = 0; j < A_DIM; ++j) y += corr[i * A_DIM + j] * v[j];
    red[i] = y * y;
    __syncthreads();
    for (int off = 64; off >= 1; off >>= 1) {
      if (i < off) red[i] += red[i + off];
      __syncthreads();
    }
    nrm = sqrtf(red[0]);
    __syncthreads();
    v[i] = y / nrm;
    __syncthreads();
  }
  if (i == 0) scal[6] = nrm;  // lambda_max (corr is PSD)
}

// --------------------------------- MLP -------------------------------------
__global__ void mlp_kernel(const float* __restrict__ X, const float* __restrict__ positions,
                           const float* __restrict__ w1, const float* __restrict__ b1,
                           const float* __restrict__ gamma, const float* __restrict__ beta,
                           const float* __restrict__ w2, const float* __restrict__ b2,
                           const float* __restrict__ w3, const float* __restrict__ b3,
                           float* __restrict__ scal) {
  __shared__ float feats[2 * A_DIM];
  __shared__ float h1[128];
  __shared__ float h2[64];
  const int i = threadIdx.x;  // 128 threads
  feats[i]         = X[(T_DIM - 1) * A_DIM + i];
  feats[A_DIM + i] = positions[i];
  __syncthreads();
  float acc = b1[i];
  for (int k = 0; k < 2 * A_DIM; ++k) acc += feats[k] * w1[k * 128 + i];
  acc = fmaxf(acc, 0.f);
  const float bnscale = 1.f / sqrtf(1.f + 1e-5f);
  h1[i] = gamma[i] * (acc * bnscale) + beta[i];
  __syncthreads();
  if (i < 64) {
    float a2 = b2[i];
    for (int k = 0; k < 128; ++k) a2 += h1[k] * w2[k * 64 + i];
    h2[i] = fmaxf(a2, 0.f);
  }
  __syncthreads();
  if (i == 0) {
    float l0 = b3[0], l1 = b3[1], l2 = b3[2];
    for (int k = 0; k < 64; ++k) {
      float h = h2[k];
      l0 += h * w3[k * 3 + 0]; l1 += h * w3[k * 3 + 1]; l2 += h * w3[k * 3 + 2];
    }
    float m = fmaxf(l0, fmaxf(l1, l2));
    float e0 = __expf(l0 - m), e1 = __expf(l1 - m), e2 = __expf(l2 - m);
    scal[10] = e2 / (e0 + e1 + e2);  // herding_severity
  }
}

// ------------------------------- finalize ----------------------------------
__global__ void final_kernel(const float* __restrict__ scal,
                             const float* __restrict__ positions,
                             float* __restrict__ out) {
  const float disp = scal[0];
  const float trend = -(scal[2] - scal[1]) / (float)(T_DIM - 1);
  float hi = trend / (disp + 1e-6f) + 0.5f;
  hi = fminf(fmaxf(hi, 0.f), 1.f);
  const float sync_risk = fminf(1.f, (scal[6] / (float)A_DIM) * scal[3]);
  float pmax = 0.f, psum = 0.f;
  for (int i = 0; i < A_DIM; ++i) {
    float p = fabsf(positions[i]); pmax = fmaxf(pmax, p); psum += p;
  }
  const float pos_div = 1.f - pmax / psum;
  const float div_loss = 1.f - sqrtf(fmaxf(scal[4] * pos_div, 0.f));
  const float hist = scal[9], rec = scal[8];
  float surge = 0.f;
  if (hist > 0.f) surge = fminf(fmaxf((rec - hist) / hist, 0.f), 1.f);
  const float pc = fminf(fmaxf((scal[5] - 0.5f) * 2.f, 0.f), 1.f);
  out[0] = hi;
  out[1] = scal[10];
  out[2] = sync_risk;
  out[3] = scal[7];
  out[4] = div_loss;
  out[5] = surge;
  out[6] = pc;
  out[7] = (hi + sync_risk + div_loss) / 3.f;
}

// ------------------------------- launcher ----------------------------------
extern "C" void kernel_launch(void* const* d_in, const int* in_sizes, int n_in,
                              void* d_out, int out_size, void* d_ws, size_t ws_size,
                              hipStream_t stream) {
  const float* X         = (const float*)d_in[0];
  const float* positions = (const float*)d_in[1];
  const float* w1 = (const float*)d_in[2];
  const float* b1 = (const float*)d_in[3];
  const float* gm = (const float*)d_in[4];
  const float* bt = (const float*)d_in[5];
  const float* w2 = (const float*)d_in[6];
  const float* b2 = (const float*)d_in[7];
  const float* w3 = (const float*)d_in[8];
  const float* b3 = (const float*)d_in[9];
  float* ws  = (float*)d_ws;
  float* out = (float*)d_out;

  float* cs_std = ws + OFF_CSSTD;
  float* r20    = ws + OFF_R20;
  float* r10    = ws + OFF_R10;
  float* S0     = ws + OFF_S0;
  float* S1     = ws + OFF_S1;
  float* S2     = ws + OFF_S2;
  float* corr   = ws + OFF_CORR;
  float* csum   = ws + OFF_CSUM;
  float* na     = ws + OFF_NA;
  float* scal   = ws + OFF_SCAL;

  csstd_kernel<<<T_DIM / 8, 256, 0, stream>>>(X, cs_std);
  colsums_kernel<<<1, 128, 0, stream>>>(X, csum, na);
  gram_kernel<0><<<64, 128, 0, stream>>>(X, S0);
  gram_kernel<1><<<64, 128, 0, stream>>>(X, S1);
  gram_kernel<2><<<64, 128, 0, stream>>>(X, S2);
  rolling_kernel<20><<<NW20, 128, 0, stream>>>(X, r20);
  rolling_kernel<10><<<NW10, 128, 0, stream>>>(X, r10);
  build_corr_kernel<<<64, 256, 0, stream>>>(S0, csum, corr);
  reduce_kernel<<<1, 256, 0, stream>>>(corr, S1, S2, na, scal);
  scalar_reduce_kernel<<<1, 256, 0, stream>>>(cs_std, r20, r10, scal);
  power_kernel<<<1, 128, 0, stream>>>(corr, scal);
  mlp_kernel<<<1, 128, 0, stream>>>(X, positions, w1, b1, gm, bt, w2, b2, w3, b3, scal);
  final_kernel<<<1, 1, 0, stream>>>(scal, positions, out);
}